// GraphConvolution_4286377361470
// MI455X (gfx1250) — compile-verified
//
#include <hip/hip_runtime.h>

typedef __attribute__((ext_vector_type(2))) float v2f;
typedef __attribute__((ext_vector_type(8))) float v8f;
typedef __attribute__((ext_vector_type(4))) int   v4i;

typedef __attribute__((address_space(1))) int  g_i32;   // global int
typedef __attribute__((address_space(3))) int  l_i32;   // LDS int
typedef __attribute__((address_space(1))) v4i  g_v4i;   // global int4
typedef __attribute__((address_space(3))) v4i  l_v4i;   // LDS int4

// ---------------- cumsum over relation axis ----------------
__global__ __launch_bounds__(256)
void cumsum_r_kernel(const float* __restrict__ w, float* __restrict__ wc,
                     int R, long plane)
{
    long i = (long)blockIdx.x * 256 + threadIdx.x;
    if (i >= plane) return;
    float s = 0.0f;
    for (int r = 0; r < R; ++r) {
        s += w[(long)r * plane + i];
        wc[(long)r * plane + i] = s;
    }
}

// ---------------- fp32 WMMA GEMM ----------------
// C[z] = A[z] (MxK, rm) * B[z] (KxN, rm) [+ bias][relu]
// BM=128, BN=64, BK=32; 256 thr = 8 waves (4x2), wave tile 32x32.
#define BM 128
#define BN 64
#define BK 32
#define LDA_S 36                  // padded LDS row stride for A (bank spread)
#define SB_PITCH (2 * BN)         // k-pair interleaved B: [k>>1][c*2 + (k&1)]
#define SA_SIZE (BM * LDA_S)
#define SB_SIZE ((BK / 2) * SB_PITCH)

#if defined(__gfx1250__) && \
    __has_builtin(__builtin_amdgcn_global_load_async_to_lds_b128) && \
    __has_builtin(__builtin_amdgcn_global_load_async_to_lds_b32)  && \
    __has_builtin(__builtin_amdgcn_s_wait_asynccnt)
#define USE_ASYNC_LDS 1
#else
#define USE_ASYNC_LDS 0
#endif

__device__ __forceinline__
void stage_tile(int tid, const float* __restrict__ A, const float* __restrict__ B,
                float* sA, float* sB, int K, int N, int bm, int bn, int k0)
{
    // A tile: 128x32 floats = 1024 float4; 4 per thread (b128 copies)
    #pragma unroll
    for (int i = 0; i < 4; ++i) {
        int idx = tid + i * 256;
        int r = idx >> 3, c4 = idx & 7;
        const float* g = A + (long)(bm + r) * K + (k0 + c4 * 4);
        float* l = sA + r * LDA_S + c4 * 4;
#if USE_ASYNC_LDS
        __builtin_amdgcn_global_load_async_to_lds_b128((g_v4i*)g, (l_v4i*)l, 0, 0);
#else
        *(float4*)l = *(const float4*)g;
#endif
    }
    // B tile: 32x64 floats; scatter into k-pair interleaved layout
    #pragma unroll
    for (int i = 0; i < 2; ++i) {
        int idx = tid + i * 256;
        int r = idx >> 4, c4 = idx & 15;
        const float* g = B + (long)(k0 + r) * N + (bn + c4 * 4);
        float* lbase = sB + (r >> 1) * SB_PITCH + (r & 1);
#if USE_ASYNC_LDS
        #pragma unroll
        for (int j = 0; j < 4; ++j) {
            __builtin_amdgcn_global_load_async_to_lds_b32(
                (g_i32*)(g + j), (l_i32*)(lbase + (c4 * 4 + j) * 2), 0, 0);
        }
#else
        float4 v = *(const float4*)g;
        lbase[(c4 * 4 + 0) * 2] = v.x;
        lbase[(c4 * 4 + 1) * 2] = v.y;
        lbase[(c4 * 4 + 2) * 2] = v.z;
        lbase[(c4 * 4 + 3) * 2] = v.w;
#endif
    }
}

__global__ __launch_bounds__(256)
void gemm_f32_wmma_kernel(const float* __restrict__ A,
                          const float* __restrict__ B,
                          float* __restrict__ C,
                          const float* __restrict__ bias,
                          int M, int N, int K, int relu,
                          long strideAz, long strideBz, long strideCz)
{
    __shared__ float sA[2 * SA_SIZE];
    __shared__ float sB[2 * SB_SIZE];

    const int tid  = threadIdx.x;
    const int lane = tid & 31;
    const int wave = tid >> 5;
    const int wm   = wave >> 1;      // 0..3
    const int wn   = wave & 1;       // 0..1
    const int l16  = lane & 15;
    const int lhi  = lane >> 4;      // 0 or 1

    const int bm = blockIdx.y * BM;
    const int bn = blockIdx.x * BN;
    const int z  = blockIdx.z;

    A += (long)z * strideAz;
    B += (long)z * strideBz;
    C += (long)z * strideCz;

    v8f acc00 = {}, acc01 = {}, acc10 = {}, acc11 = {};

    stage_tile(tid, A, B, sA, sB, K, N, bm, bn, 0);
    int buf = 0;

    for (int k0 = 0; k0 < K; k0 += BK) {
#if USE_ASYNC_LDS
        __builtin_amdgcn_s_wait_asynccnt(0);   // my async stores to LDS done
#endif
        __syncthreads();                       // tile 'buf' visible to all waves

        if (k0 + BK < K)
            stage_tile(tid, A, B, sA + (buf ^ 1) * SA_SIZE, sB + (buf ^ 1) * SB_SIZE,
                       K, N, bm, bn, k0 + BK); // prefetch next tile (overlaps)

        const float* cA = sA + buf * SA_SIZE;
        const float* cB = sB + buf * SB_SIZE;

        #pragma unroll
        for (int kk = 0; kk < BK; kk += 4) {
            const int ak = kk + lhi * 2;
            v2f a0 = *(const v2f*)(cA + (wm * 32 + l16) * LDA_S + ak);
            v2f a1 = *(const v2f*)(cA + (wm * 32 + 16 + l16) * LDA_S + ak);
            v2f b0 = *(const v2f*)(cB + (ak >> 1) * SB_PITCH + (wn * 32 + l16) * 2);
            v2f b1 = *(const v2f*)(cB + (ak >> 1) * SB_PITCH + (wn * 32 + 16 + l16) * 2);

            acc00 = __builtin_amdgcn_wmma_f32_16x16x4_f32(false, a0, false, b0,
                                                          (short)0, acc00, false, false);
            acc01 = __builtin_amdgcn_wmma_f32_16x16x4_f32(false, a0, false, b1,
                                                          (short)0, acc01, false, false);
            acc10 = __builtin_amdgcn_wmma_f32_16x16x4_f32(false, a1, false, b0,
                                                          (short)0, acc10, false, false);
            acc11 = __builtin_amdgcn_wmma_f32_16x16x4_f32(false, a1, false, b1,
                                                          (short)0, acc11, false, false);
        }
        __syncthreads();                       // all waves done reading 'buf'
        buf ^= 1;
    }

    // ---- epilogue ----
    const int col0 = bn + wn * 32 + l16;
    const int col1 = col0 + 16;
    const float bv0 = bias ? bias[col0] : 0.0f;
    const float bv1 = bias ? bias[col1] : 0.0f;
    const int row0 = bm + wm * 32 + lhi * 8;
    const int row1 = row0 + 16;

    #pragma unroll
    for (int i = 0; i < 8; ++i) {
        float v00 = acc00[i] + bv0;
        float v01 = acc01[i] + bv1;
        float v10 = acc10[i] + bv0;
        float v11 = acc11[i] + bv1;
        if (relu) {
            v00 = fmaxf(v00, 0.0f); v01 = fmaxf(v01, 0.0f);
            v10 = fmaxf(v10, 0.0f); v11 = fmaxf(v11, 0.0f);
        }
        C[(long)(row0 + i) * N + col0] = v00;
        C[(long)(row0 + i) * N + col1] = v01;
        C[(long)(row1 + i) * N + col0] = v10;
        C[(long)(row1 + i) * N + col1] = v11;
    }
}

// ---------------- launch ----------------
extern "C" void kernel_launch(void* const* d_in, const int* in_sizes, int n_in,
                              void* d_out, int out_size, void* d_ws, size_t ws_size,
                              hipStream_t stream)
{
    (void)in_sizes; (void)n_in; (void)out_size; (void)ws_size;

    constexpr int U = 4096, V = 4096, R = 5, I = 2048, H = 512;

    const float* u_feat    = (const float*)d_in[0];
    const float* v_feat    = (const float*)d_in[1];
    const float* support   = (const float*)d_in[2];
    const float* support_t = (const float*)d_in[3];
    const float* u_weight  = (const float*)d_in[4];
    const float* u_bias    = (const float*)d_in[5];

    float* out_zu = (float*)d_out;
    float* out_zv = out_zu + (long)U * H;

    float* w_cum = (float*)d_ws;
    float* tmp_u = w_cum + (long)R * I * H;
    float* tmp_v = tmp_u + (long)R * U * H;

    const long plane = (long)I * H;

    {
        dim3 grid((unsigned)((plane + 255) / 256));
        cumsum_r_kernel<<<grid, 256, 0, stream>>>(u_weight, w_cum, R, plane);
    }
    {
        dim3 grid(H / BN, U / BM, R);
        gemm_f32_wmma_kernel<<<grid, 256, 0, stream>>>(
            u_feat, w_cum, tmp_u, nullptr, U, H, I, 0,
            0L, (long)I * H, (long)U * H);
    }
    {
        dim3 grid(H / BN, V / BM, R);
        gemm_f32_wmma_kernel<<<grid, 256, 0, stream>>>(
            v_feat, w_cum, tmp_v, nullptr, V, H, I, 0,
            0L, (long)I * H, (long)V * H);
    }
    {
        dim3 grid(H / BN, U / BM, 1);
        gemm_f32_wmma_kernel<<<grid, 256, 0, stream>>>(
            support, tmp_v, out_zu, u_bias, U, H, R * V, 1,
            0L, 0L, 0L);
    }
    {
        dim3 grid(H / BN, V / BM, 1);
        gemm_f32_wmma_kernel<<<grid, 256, 0, stream>>>(
            support_t, tmp_u, out_zv, u_bias, V, H, R * U, 1,
            0L, 0L, 0L);
    }
}